// Network_62878321214134
// MI455X (gfx1250) — compile-verified
//
#include <hip/hip_runtime.h>

#define L      128
#define XW     256          // X row = [z(128) | h(128)]
#define MAXIT  10

typedef __attribute__((ext_vector_type(2))) float v2f;
typedef __attribute__((ext_vector_type(8))) float v8f;

// ---------------------------------------------------------------------------
// Stage a K x 64 strip of row-major B (ldb cols) into LDS, transposed:
// BT[c][k] = B[k][bn0+c], row stride KP = K+2 (even pad: conflict-free staging
// stores, and 8B-aligned ds_load_b64 in the compute phase).
template <int K>
__device__ __forceinline__ void stage_BT(const float* __restrict__ B, int ldb,
                                         int bn0, float* __restrict__ BT) {
  const int KP = K + 2;
  const int t = threadIdx.x;
#pragma unroll
  for (int i = 0; i < K / 16; ++i) {          // 256 thr * (K/16) * 4 = 64*K elems
    int id = t + i * 256;
    int k  = id >> 4;
    int c4 = (id & 15) * 4;
    float4 v = *(const float4*)(B + (size_t)k * ldb + bn0 + c4);
    BT[(c4 + 0) * KP + k] = v.x;
    BT[(c4 + 1) * KP + k] = v.y;
    BT[(c4 + 2) * KP + k] = v.z;
    BT[(c4 + 3) * KP + k] = v.w;
  }
}

// One wave computes a 16x64 strip: 4 accumulators, A frag reused 4x per k-step.
// A frag (16x4 f32): lanes0-15 M=0..15 K={k,k+1}; lanes16-31 K={k+2,k+3}.
template <int K>
__device__ __forceinline__ void wmma_strip(const float* __restrict__ A, int lda,
                                           int m0, const float* __restrict__ BT,
                                           v8f acc[4]) {
  const int KP = K + 2;
  const int lane = threadIdx.x & 31;
  const int half = lane >> 4;
  const float* arow = A + (size_t)(m0 + (lane & 15)) * lda + half * 2;
  const float* brow = BT + (lane & 15) * KP + half * 2;
#pragma unroll 2
  for (int k = 0; k < K; k += 4) {
    v2f a = *(const v2f*)(arow + k);          // global_load_b64 (L2-resident)
#pragma unroll
    for (int nt = 0; nt < 4; ++nt) {
      v2f b = *(const v2f*)(brow + nt * 16 * KP + k);   // ds_load_b64
      acc[nt] = __builtin_amdgcn_wmma_f32_16x16x4_f32(false, a, false, b,
                                                      (short)0, acc[nt],
                                                      false, false);
    }
  }
}

// P[:,0:128] = X @ W1 ; P[:,128:256] = X @ W2   (no bias/relu here)
__global__ void __launch_bounds__(256)
gemm_P(const float* __restrict__ X, const float* __restrict__ W1,
       const float* __restrict__ W2, float* __restrict__ P, int N) {
  __shared__ float BT[64 * (XW + 2)];
  const int strip = blockIdx.y;                       // 0..3 -> out cols strip*64
  const float* B = (strip < 2) ? W1 : W2;
  stage_BT<XW>(B, L, (strip & 1) * 64, BT);
  __syncthreads();
  const int wave = threadIdx.x >> 5, lane = threadIdx.x & 31;
  const int tm = blockIdx.x * 8 + wave;
  if (tm * 16 >= N) return;                           // wave-uniform, after barrier
  v8f acc[4] = {};
  wmma_strip<XW>(X, XW, tm * 16, BT, acc);
  const int mbase = tm * 16 + (lane >> 4) * 8;
  const int c0 = strip * 64 + (lane & 15);
#pragma unroll
  for (int nt = 0; nt < 4; ++nt)
#pragma unroll
    for (int r = 0; r < 8; ++r)
      P[(size_t)(mbase + r) * XW + c0 + nt * 16] = acc[nt][r];
}

// h = relu(AGG @ W_out + b_out), written into X[:,128:256] for the next iter
__global__ void __launch_bounds__(256)
gemm_H(const float* __restrict__ AGG, const float* __restrict__ Wout,
       const float* __restrict__ bout, float* __restrict__ Xo, int N) {
  __shared__ float BT[64 * (L + 2)];
  const int strip = blockIdx.y;                       // 0..1
  stage_BT<L>(Wout, L, strip * 64, BT);
  __syncthreads();
  const int wave = threadIdx.x >> 5, lane = threadIdx.x & 31;
  const int tm = blockIdx.x * 8 + wave;
  if (tm * 16 >= N) return;
  v8f acc[4] = {};
  wmma_strip<L>(AGG, L, tm * 16, BT, acc);
  const int mbase = tm * 16 + (lane >> 4) * 8;
  const int c0 = strip * 64 + (lane & 15);
#pragma unroll
  for (int nt = 0; nt < 4; ++nt) {
    const int col = c0 + nt * 16;
    const float bias = bout[col];
#pragma unroll
    for (int r = 0; r < 8; ++r)
      Xo[(size_t)(mbase + r) * XW + L + col] = fmaxf(acc[nt][r] + bias, 0.0f);
  }
}

// ---------------------------------------------------------------------------
__global__ void init_xh(float* __restrict__ X, int N) {   // zero h half of X
  int t = blockIdx.x * blockDim.x + threadIdx.x;
  if (t < N * L) X[(size_t)(t >> 7) * XW + L + (t & 127)] = 0.0f;
}

__global__ void fill_zero(int* __restrict__ p, long n) {
  long t = (long)blockIdx.x * blockDim.x + threadIdx.x;
  if (t < n) p[t] = 0;
}

// z = relu(pos*Wenc[0,:] + reach*Wenc[1,:] + benc) -> X[:,0:128]
__global__ void build_x(const float* __restrict__ pos, const float* __restrict__ reach,
                        const float* __restrict__ Wenc, const float* __restrict__ benc,
                        float* __restrict__ X, int N) {
  int t = blockIdx.x * blockDim.x + threadIdx.x;
  if (t >= N * L) return;
  int n = t >> 7, j = t & 127;
  float z = fmaf(pos[n], Wenc[j], fmaf(reach[n], Wenc[L + j], benc[j]));
  X[(size_t)n * XW + j] = fmaxf(z, 0.0f);
}

// per edge: m = relu(P1[src]+P2[dst]+b_msg); scatter-max into agg[dst]
// (all values >= 0 -> IEEE bits order like ints -> integer atomicMax is exact)
__global__ void __launch_bounds__(256)
edge_msg(const float* __restrict__ P, const int* __restrict__ src,
         const int* __restrict__ dst, const float* __restrict__ bmsg,
         int* __restrict__ agg, int E) {
  int t = blockIdx.x * blockDim.x + threadIdx.x;
  int e = t >> 5;
  if (e >= E) return;
  int j = (t & 31) * 4;
  int s = src[e], d = dst[e];
  float4 p1 = *(const float4*)(P + (size_t)s * XW + j);
  float4 p2 = *(const float4*)(P + (size_t)d * XW + L + j);
  float4 b  = *(const float4*)(bmsg + j);
  int* a = agg + (size_t)d * L + j;
  atomicMax(a + 0, __float_as_int(fmaxf(p1.x + p2.x + b.x, 0.0f)));
  atomicMax(a + 1, __float_as_int(fmaxf(p1.y + p2.y + b.y, 0.0f)));
  atomicMax(a + 2, __float_as_int(fmaxf(p1.z + p2.z + b.z, 0.0f)));
  atomicMax(a + 3, __float_as_int(fmaxf(p1.w + p2.w + b.w, 0.0f)));
}

// q1[n] = h[n] . Wdec[0:128], q2[n] = h[n] . Wdec[128:256]  (one wave per node)
__global__ void __launch_bounds__(256)
qdec(const float* __restrict__ X, const float* __restrict__ Wdec,
     float* __restrict__ q1, float* __restrict__ q2, int N) {
  int n = blockIdx.x * 8 + (threadIdx.x >> 5);
  int lane = threadIdx.x & 31;
  if (n >= N) return;
  float4 h  = *(const float4*)(X + (size_t)n * XW + L + lane * 4);
  float4 w1 = *(const float4*)(Wdec + lane * 4);
  float4 w2 = *(const float4*)(Wdec + L + lane * 4);
  float s1 = h.x * w1.x + h.y * w1.y + h.z * w1.z + h.w * w1.w;
  float s2 = h.x * w2.x + h.y * w2.y + h.z * w2.z + h.w * w2.w;
#pragma unroll
  for (int off = 16; off; off >>= 1) {
    s1 += __shfl_down(s1, off, 32);
    s2 += __shfl_down(s2, off, 32);
  }
  if (lane == 0) { q1[n] = s1; q2[n] = s2; }
}

// alpha = sigmoid(q1[src]+q2[dst]+b_dec); store pred; update node maxes
__global__ void __launch_bounds__(256)
alpha_k(const float* __restrict__ q1, const float* __restrict__ q2,
        const int* __restrict__ src, const int* __restrict__ dst,
        const float* __restrict__ bdec, float* __restrict__ pred,
        int* __restrict__ nodemax, int* __restrict__ mxd, int E) {
  int e = blockIdx.x * blockDim.x + threadIdx.x;
  if (e >= E) return;
  int s = src[e], d = dst[e];
  float a = 1.0f / (1.0f + __expf(-(q1[s] + q2[d] + bdec[0])));
  pred[e] = a;
  int ai = __float_as_int(a);                        // sigmoid > 0
  atomicMax(nodemax + s, ai);
  atomicMax(nodemax + d, ai);
  atomicMax(mxd + d, ai);
}

__global__ void reach_k(const int* __restrict__ nodemax, float* __restrict__ reach,
                        float* __restrict__ out_reach, int N) {
  int n = blockIdx.x * blockDim.x + threadIdx.x;
  if (n >= N) return;
  float r = (__int_as_float(nodemax[n]) >= 0.4f) ? 1.0f : 0.0f;
  reach[n] = r;
  out_reach[n] = r;
}

__global__ void init_best(int* __restrict__ best, int N, int E) {
  int n = blockIdx.x * blockDim.x + threadIdx.x;
  if (n < N) best[n] = E;
}

__global__ void parent_a(const float* __restrict__ alpha, const int* __restrict__ mxd,
                         const int* __restrict__ dst, int* __restrict__ best, int E) {
  int e = blockIdx.x * blockDim.x + threadIdx.x;
  if (e >= E) return;
  int d = dst[e];
  if (alpha[e] == __int_as_float(mxd[d]))            // alpha >= 0 always
    atomicMin(best + d, e);
}

__global__ void parent_b(const int* __restrict__ best, const int* __restrict__ src,
                         float* __restrict__ out_par, int N, int E) {
  int n = blockIdx.x * blockDim.x + threadIdx.x;
  if (n >= N) return;
  int b = best[n];
  out_par[n] = (b < E) ? (float)src[b] : (float)n;
}

// ---------------------------------------------------------------------------
extern "C" void kernel_launch(void* const* d_in, const int* in_sizes, int n_in,
                              void* d_out, int out_size, void* d_ws, size_t ws_size,
                              hipStream_t stream) {
  const float* pos  = (const float*)d_in[0];
  const float* s    = (const float*)d_in[1];
  const int*   ei   = (const int*)d_in[2];
  const float* Wenc = (const float*)d_in[3];
  const float* benc = (const float*)d_in[4];
  const float* W1   = (const float*)d_in[5];
  const float* W2   = (const float*)d_in[6];
  const float* bmsg = (const float*)d_in[7];
  const float* Wout = (const float*)d_in[8];
  const float* bout = (const float*)d_in[9];
  const float* Wdec = (const float*)d_in[10];
  const float* bdec = (const float*)d_in[11];

  const int N = in_sizes[0];
  const int E = in_sizes[2] / 2;
  const int* src = ei;
  const int* dst = ei + E;

  // workspace layout (all fp32/int32, 16B-aligned since N*4 is)
  float* X    = (float*)d_ws;                    // N*256
  float* P    = X + (size_t)N * XW;              // N*256
  float* AGG  = P + (size_t)N * XW;              // N*128  (int-bit max accum)
  int* nodemax = (int*)(AGG + (size_t)N * L);    // N   } contiguous with AGG:
  int* mxd     = nodemax + N;                    // N   } zeroed in one fill
  float* q1   = (float*)(mxd + N);               // N
  float* q2   = q1 + N;                          // N
  float* reach = q2 + N;                         // N
  int* best   = (int*)(reach + N);               // N

  float* preds     = (float*)d_out;                    // MAXIT*E
  float* out_reach = preds + (size_t)MAXIT * E;        // N
  float* out_par   = out_reach + N;                    // N

  const int mblk = ((N / 16) + 7) / 8;           // blocks of 8 M-tiles
  init_xh<<<(N * L + 255) / 256, 256, 0, stream>>>(X, N);

  for (int it = 0; it < MAXIT; ++it) {
    long zn = (long)N * (L + 2);                 // AGG + nodemax + mxd
    fill_zero<<<(unsigned)((zn + 255) / 256), 256, 0, stream>>>((int*)AGG, zn);
    build_x<<<(N * L + 255) / 256, 256, 0, stream>>>(pos, it == 0 ? s : reach,
                                                     Wenc, benc, X, N);
    gemm_P<<<dim3(mblk, 4), 256, 0, stream>>>(X, W1, W2, P, N);
    edge_msg<<<(E * 32 + 255) / 256, 256, 0, stream>>>(P, src, dst, bmsg,
                                                       (int*)AGG, E);
    gemm_H<<<dim3(mblk, 2), 256, 0, stream>>>(AGG, Wout, bout, X, N);
    qdec<<<(N + 7) / 8, 256, 0, stream>>>(X, Wdec, q1, q2, N);
    alpha_k<<<(E + 255) / 256, 256, 0, stream>>>(q1, q2, src, dst, bdec,
                                                 preds + (size_t)it * E,
                                                 nodemax, mxd, E);
    reach_k<<<(N + 255) / 256, 256, 0, stream>>>(nodemax, reach, out_reach, N);
  }

  init_best<<<(N + 255) / 256, 256, 0, stream>>>(best, N, E);
  parent_a<<<(E + 255) / 256, 256, 0, stream>>>(preds + (size_t)(MAXIT - 1) * E,
                                                mxd, dst, best, E);
  parent_b<<<(N + 255) / 256, 256, 0, stream>>>(best, src, out_par, N, E);
}